// BrockHommes_34213709480312
// MI455X (gfx1250) — compile-verified
//
#include <hip/hip_runtime.h>
#include <math.h>

// ---------------------------------------------------------------------------
// Brock-Hommes scan: B independent serial chains of length T.
//   per step: 4-way softmax over row-local logits -> mean -> AR(3) update.
// One thread per row. Epsilon tiles staged global->LDS with CDNA5 async DMA
// (GLOBAL_LOAD_ASYNC_TO_LDS_B32, tracked by ASYNCcnt), double buffered.
// Outputs stored as per-thread contiguous float4 (full 128B line per lane).
// ---------------------------------------------------------------------------

#define ROWS 128            // threads per block == rows per block (4 wave32s)
#define TT   32             // time-chunk staged per LDS buffer
#define LSTRIDE (TT + 1)    // 33-float row stride -> conflict-free LDS reads

typedef __attribute__((address_space(1))) int* as1_int;   // global
typedef __attribute__((address_space(3))) int* as3_int;   // LDS

#if __has_builtin(__builtin_amdgcn_s_wait_asynccnt)
#define WAIT_ASYNC(n) __builtin_amdgcn_s_wait_asynccnt(n)
#else
#define WAIT_ASYNC(n) asm volatile("s_wait_asynccnt %0" :: "i"(n) : "memory")
#endif

#if __has_builtin(__builtin_amdgcn_global_load_async_to_lds_b32)
#define HAVE_ASYNC_LDS 1
#else
#define HAVE_ASYNC_LDS 0
#warning "CDNA5 probe: __builtin_amdgcn_global_load_async_to_lds_b32 NOT available"
#endif

__device__ __forceinline__ void async_ld_b32(const float* gsrc, float* ldst) {
#if HAVE_ASYNC_LDS
  __builtin_amdgcn_global_load_async_to_lds_b32(
      (as1_int)const_cast<float*>(gsrc), (as3_int)ldst,
      /*imm offset*/0, /*cpol*/0);
#else
  *ldst = __builtin_nontemporal_load(gsrc);   // sync fallback, still correct
#endif
}

struct BHParams {
  float beta, g0, g1, g2, g3, b0, b1, b2, b3, sigma;
  float R, invR, sinvR;
};

// one recurrence step; returns x_t, shifts carry
__device__ __forceinline__ float bh_step(float ep, float& x1, float& x2,
                                         float& x3, const BHParams& p) {
  const float Rx2 = p.R * x2;
  const float a   = p.beta * (x1 - Rx2);
  const float e0  = a * (fmaf(p.g0, x3, p.b0) - Rx2);
  const float e1  = a * (fmaf(p.g1, x3, p.b1) - Rx2);
  const float e2  = a * (fmaf(p.g2, x3, p.b2) - Rx2);
  const float e3  = a * (fmaf(p.g3, x3, p.b3) - Rx2);
  const float m   = fmaxf(fmaxf(e0, e1), fmaxf(e2, e3));
  const float w0  = __expf(e0 - m);
  const float w1  = __expf(e1 - m);
  const float w2  = __expf(e2 - m);
  const float w3  = __expf(e3 - m);
  const float s   = (w0 + w1) + (w2 + w3);
  float num = w0 * fmaf(p.g0, x1, p.b0);
  num = fmaf(w1, fmaf(p.g1, x1, p.b1), num);
  num = fmaf(w2, fmaf(p.g2, x1, p.b2), num);
  num = fmaf(w3, fmaf(p.g3, x1, p.b3), num);
#if __has_builtin(__builtin_amdgcn_rcpf)
  const float mean = num * __builtin_amdgcn_rcpf(s);
#else
  const float mean = num / s;
#endif
  const float xt = fmaf(ep, p.sinvR, mean * p.invR);
  x3 = x2; x2 = x1; x1 = xt;
  return xt;
}

__global__ __launch_bounds__(ROWS, 1)
void BrockHommes_scan_kernel(const float* __restrict__ theta,
                             const float* __restrict__ eps,
                             float* __restrict__ out,
                             int B, int T) {
  __shared__ float tile[2][ROWS * LSTRIDE];

  const int tid  = threadIdx.x;
  const int row0 = blockIdx.x * ROWS;
  const int row  = row0 + tid;
  const bool act = (row < B);
  const int  rc  = act ? row : (B > 0 ? B - 1 : 0);

  // per-row parameters (11 floats), held in registers for the whole scan
  const float* th = theta + (size_t)rc * 11;
  BHParams p;
  p.beta = th[0];
  p.g0 = th[1]; p.g1 = th[2]; p.g2 = th[3]; p.g3 = th[4];
  p.b0 = th[5]; p.b1 = th[6]; p.b2 = th[7]; p.b3 = th[8];
  p.sigma = th[9];
  p.R     = 1.0f + th[10];
  p.invR  = 1.0f / p.R;
  p.sinvR = p.sigma * p.invR;

  const int nch = T / TT;

  // Cooperative coalesced async stage of one ROWSxTT epsilon tile:
  // flat e = tid + it*ROWS; each wave32 covers one row's 32 consecutive
  // floats (128B contiguous global segment) per iteration.
  auto issue = [&](int k, int buf) {
    const size_t t0 = (size_t)k * TT;
#pragma unroll
    for (int it = 0; it < TT; ++it) {            // ROWS*TT / ROWS == TT
      const int e = tid + it * ROWS;
      const int r = e / TT;
      const int c = e % TT;
      int gr = row0 + r;
      gr = (gr < B) ? gr : (B - 1);
      async_ld_b32(eps + (size_t)gr * T + t0 + c,
                   &tile[buf][r * LSTRIDE + c]);
    }
  };

  float x1 = 0.0f, x2 = 0.0f, x3 = 0.0f;
  if (nch > 0) issue(0, 0);

  for (int k = 0; k < nch; ++k) {
    const int buf = k & 1;
    if (k + 1 < nch) {
      issue(k + 1, buf ^ 1);     // prefetch next chunk into other buffer
      WAIT_ASYNC(TT);            // <=TT outstanding => chunk k has landed
    } else {
      WAIT_ASYNC(0);
    }
    __syncthreads();             // publish tile across all waves

    float xb[TT];
    const float* myrow = &tile[buf][tid * LSTRIDE];
#pragma unroll
    for (int c = 0; c < TT; ++c) {
      xb[c] = bh_step(myrow[c], x1, x2, x3, p);
    }

    if (act) {
      float* orow = out + (size_t)row * T + (size_t)k * TT;
      if ((T & 3) == 0) {        // 16B-aligned per-thread contiguous stores
#pragma unroll
        for (int q = 0; q < TT / 4; ++q) {
          float4 v = make_float4(xb[4*q], xb[4*q+1], xb[4*q+2], xb[4*q+3]);
          *reinterpret_cast<float4*>(orow + 4*q) = v;
        }
      } else {
#pragma unroll
        for (int c = 0; c < TT; ++c) orow[c] = xb[c];
      }
    }
    __syncthreads();             // buf[k&1] free for reuse at chunk k+2
  }

  // tail when T is not a multiple of TT (not hit for T=4096)
  for (int t = nch * TT; t < T; ++t) {
    const float ep = act ? eps[(size_t)row * T + t] : 0.0f;
    const float xt = bh_step(ep, x1, x2, x3, p);
    if (act) out[(size_t)row * T + t] = xt;
  }
}

extern "C" void kernel_launch(void* const* d_in, const int* in_sizes, int n_in,
                              void* d_out, int out_size, void* d_ws, size_t ws_size,
                              hipStream_t stream) {
  (void)n_in; (void)out_size; (void)d_ws; (void)ws_size;
  const float* theta = (const float*)d_in[0];   // (B, 11) fp32
  const float* eps   = (const float*)d_in[1];   // (B, T)  fp32
  float* out = (float*)d_out;                   // (B, T)  fp32

  const int B = in_sizes[0] / 11;
  const int T = (B > 0) ? (in_sizes[1] / B) : 0;
  if (B <= 0 || T <= 0) return;

  const int blocks = (B + ROWS - 1) / ROWS;
  hipLaunchKernelGGL(BrockHommes_scan_kernel, dim3(blocks), dim3(ROWS), 0,
                     stream, theta, eps, out, B, T);
}